// VoronoiDecoder_39350490366154
// MI455X (gfx1250) — compile-verified
//
#include <hip/hip_runtime.h>
#include <hip/hip_bf16.h>
#include <math.h>

// ---------------- constants from the reference ----------------
#define S_SEEDS      256
#define EPS_F        1e-8f
#define W_MIN_F      0.005f
#define BETA_F       0.02f
#define RAW_TEMP_INV 0.2f          // 1/5.0
#define H_MIN_F      0.5f
#define H_MAX_F      2.0f
#define J_WIDTH_BONUS 0.15f
#define J_KEFF_K0    3.0f
#define J_KEFF_S_INV (1.0f/0.35f)
#define J_TRIPLE_LAMBDA 0.15f
#define SOFTMIN_TAU  0.002f
#define L2E          1.4426950408889634f   // log2(e)
#define LN2_F        0.6931471805599453f

typedef __attribute__((ext_vector_type(2))) float v2f;
typedef __attribute__((ext_vector_type(8))) float v8f;

// fast-path device math: raw v_exp_f32 / v_log_f32 / v_sqrt_f32 / v_rcp_f32 / v_med3_f32
__device__ __forceinline__ float fexp2(float x) { return __builtin_amdgcn_exp2f(x); }
__device__ __forceinline__ float flog2(float x) { return __builtin_amdgcn_logf(x); }
__device__ __forceinline__ float fsqrt(float x) { return __builtin_amdgcn_sqrtf(x); }
__device__ __forceinline__ float frcp (float x) { return __builtin_amdgcn_rcpf(x); }
__device__ __forceinline__ float fclamp_lo(float x) {
    // single v_med3_f32: clamp to [EPS, 3e38] without canonicalize pre-ops
    return __builtin_amdgcn_fmed3f(x, EPS_F, 3.0e38f);
}
__device__ __forceinline__ float sigm (float x) { return frcp(1.0f + fexp2(-x * L2E)); }

// ---------------------------------------------------------------------------
// Setup: fold rotation/scale into quadratic-form coefficients so the distance
// matrix becomes a K=6 GEMM:  d^2[q,s] = f(q) . g(s)
//   f(q) = [u^2, uv, v^2, u, v, 1, 0, 0]
//   g(s) = [A,   B,  C,  g3, g4, g5+EPS, 0, 0]
// coeff stored feature-major (coeff[f*256+s]); h/gates interleaved in hg[2s].
// ---------------------------------------------------------------------------
__global__ __launch_bounds__(256) void voro_setup(
    const float* __restrict__ seeds, const float* __restrict__ h_raw,
    const float* __restrict__ theta, const float* __restrict__ a_raw,
    const float* __restrict__ seed_gates,
    float* __restrict__ coeff, float* __restrict__ hg)
{
    int s = threadIdx.x;
    float sx = seeds[2*s], sy = seeds[2*s+1];
    float th = theta[s];
    float c = cosf(th), sn = sinf(th);
    float sc  = expf(a_raw[s]);
    float sc2 = sc * sc;
    float is2 = 1.0f / sc2;
    float A = c*c*is2 + sn*sn*sc2;
    float B = 2.0f*c*sn*(is2 - sc2);
    float C = sn*sn*is2 + c*c*sc2;
    coeff[0*S_SEEDS + s] = A;
    coeff[1*S_SEEDS + s] = B;
    coeff[2*S_SEEDS + s] = C;
    coeff[3*S_SEEDS + s] = -(2.0f*A*sx + B*sy);
    coeff[4*S_SEEDS + s] = -(B*sx + 2.0f*C*sy);
    coeff[5*S_SEEDS + s] = A*sx*sx + B*sx*sy + C*sy*sy + EPS_F;
    coeff[6*S_SEEDS + s] = 0.0f;
    coeff[7*S_SEEDS + s] = 0.0f;
    hg[2*s + 0] = H_MIN_F + (H_MAX_F - H_MIN_F) * (1.0f / (1.0f + expf(-h_raw[s])));
    hg[2*s + 1] = 1.0f / (1.0f + expf(-seed_gates[s]));
}

// ---------------------------------------------------------------------------
// Main: one wave (32 lanes) per block, 32 queries per block.
// Phase A: V_WMMA_F32_16X16X4_F32 sweep -> d[32 queries][256 seeds] in LDS
//          (row stride 257 floats -> bank-conflict-free column walk in phase B)
// Phase B: lane q streams its 256 distances: fused online-softmax
//          (sum e, sum e^2, sum h*e, sum g*e) + branch-free sorted top-8,
//          with the top-8 list kept in t = -d/beta domain (canonical values).
// ---------------------------------------------------------------------------
__global__ __launch_bounds__(32) void voro_main(
    const float* __restrict__ uv, const float* __restrict__ seeds,
    const float* __restrict__ w_raw, const float* __restrict__ coeff,
    const float* __restrict__ hg, float* __restrict__ out)
{
    __shared__ float sd[32 * 257];
    __shared__ v2f   shg[S_SEEDS];     // {h, gate} pairs -> one b64 broadcast

    const int lane   = threadIdx.x;          // 0..31
    const int m      = lane & 15;            // row within 16x16 tile
    const bool hi    = lane >= 16;
    const int qblock = blockIdx.x * 32;

    {
        const v2f* hg2 = (const v2f*)hg;
        for (int i = lane; i < S_SEEDS; i += 32) shg[i] = hg2[i];
    }

    // Prefetch next block's queries into GL2 (gfx1250 global_prefetch path).
    if (blockIdx.x + 1 < gridDim.x)
        __builtin_prefetch(uv + 2 * (qblock + 32), 0, 0);

    // ---------------- Phase A: WMMA distance tiles ----------------
    // A-matrix (16x4 f32): VGPR0 = K0|K2, VGPR1 = K1|K3 (split at lane 16)
    v2f a0[2], a1[2];
    const v2f* uv2 = (const v2f*)uv;
#pragma unroll
    for (int mt = 0; mt < 2; ++mt) {
        const v2f p = uv2[qblock + mt * 16 + m];
        const float u = p.x, v = p.y;
        a0[mt].x = hi ? v * v : u * u;     // K2 = v^2   | K0 = u^2
        a0[mt].y = hi ? u     : u * v;     // K3 = u     | K1 = u*v
        a1[mt].x = hi ? 0.0f  : v;         // K6 = 0     | K4 = v
        a1[mt].y = hi ? 0.0f  : 1.0f;      // K7 = 0     | K5 = 1
    }

    for (int t = 0; t < 16; ++t) {
        const int n = t * 16 + m;          // global seed column
        v2f b0, b1;                        // B-matrix (4x16): VGPR0 = K0|K2, VGPR1 = K1|K3
        b0.x = coeff[(hi ? 2 : 0) * S_SEEDS + n];
        b0.y = coeff[(hi ? 3 : 1) * S_SEEDS + n];
        b1.x = coeff[(hi ? 6 : 4) * S_SEEDS + n];
        b1.y = coeff[(hi ? 7 : 5) * S_SEEDS + n];

#pragma unroll
        for (int mt = 0; mt < 2; ++mt) {
            v8f acc = {};
            acc = __builtin_amdgcn_wmma_f32_16x16x4_f32(
                      false, a0[mt], false, b0, (short)0, acc, false, false);
            acc = __builtin_amdgcn_wmma_f32_16x16x4_f32(
                      false, a1[mt], false, b1, (short)0, acc, false, false);

            // D layout: VGPR r holds rows r (lanes 0-15) / r+8 (lanes 16-31)
#pragma unroll
            for (int r = 0; r < 8; ++r) {
                const int row = mt * 16 + (hi ? r + 8 : r);
                sd[row * 257 + n] = fsqrt(fclamp_lo(acc[r]));
            }
        }
    }
    __syncthreads();

    // ---------------- Phase B: per-query reduction ----------------
    const float NEGK = -1.0f / BETA_F;       // -50
    float m_run = -3.0e38f;
    float sum = 0.0f, sum2 = 0.0f, hs = 0.0f, gs = 0.0f;
    // top-8 list in t-domain, sorted descending (largest t == smallest d)
    float tk[8];
#pragma unroll
    for (int j = 0; j < 8; ++j) tk[j] = -3.0e38f;
    int i1 = 0, i2 = 0;

#pragma unroll 4
    for (int s = 0; s < S_SEEDS; ++s) {
        const float d  = sd[lane * 257 + s];
        const v2f  hgv = shg[s];             // broadcast b64

        // online softmax over t = -d/beta, tracking sum e, sum e^2, h.e, g.e
        const float t  = NEGK * d;
        const float mn = fmaxf(m_run, t);
        const float sc = fexp2((m_run - mn) * L2E);
        const float e  = fexp2((t - mn) * L2E);
        sum  = fmaf(sum,  sc,      e);
        sum2 = fmaf(sum2, sc * sc, e * e);
        hs   = fmaf(hs,   sc, hgv.x * e);
        gs   = fmaf(gs,   sc, hgv.y * e);
        m_run = mn;

        // top-2 indices (before list mutates)
        const bool c0 = t > tk[0];
        const bool c1 = t > tk[1];
        i2 = c0 ? i1 : (c1 ? s : i2);
        i1 = c0 ? s  : i1;

        // branch-free sorted insert (keep 8 largest t): max/min carry chain
        float carry = t;
#pragma unroll
        for (int j = 0; j < 8; ++j) {
            const float nv = fmaxf(tk[j], carry);
            carry = fminf(tk[j], carry);
            tk[j] = nv;
        }
    }

    // ---------------- per-query epilogue ----------------
    // convert back: d_j = -beta * t_j
    const float d3md1 = BETA_F * (tk[0] - tk[2]);   // d3 - d1  (>= 0)

    // W[i1,i2] computed on the fly (w_raw stays hot in L2)
    const float s1x = seeds[2*i1], s1y = seeds[2*i1+1];
    const float s2x = seeds[2*i2], s2y = seeds[2*i2+1];
    const float pdx = s1x - s2x, pdy = s1y - s2y;
    const float pd  = fsqrt(pdx*pdx + pdy*pdy);
    const float wmax = fmaxf(0.8f * pd, W_MIN_F + EPS_F);
    const float g12 = W_MIN_F + (wmax - W_MIN_F) * sigm(w_raw[i1*S_SEEDS + i2] * RAW_TEMP_INV);
    const float g21 = W_MIN_F + (wmax - W_MIN_F) * sigm(w_raw[i2*S_SEEDS + i1] * RAW_TEMP_INV);
    const float w_pair = 0.5f * (g12 + g21);

    // k_eff bonus:  sum p^2 = sum2 / sum^2
    const float rsum  = frcp(sum);
    const float pp    = sum2 * rsum * rsum;
    const float k_eff = frcp(pp + EPS_F);
    const float bonus = J_WIDTH_BONUS * sigm((k_eff - J_KEFF_K0) * J_KEFF_S_INV);

    // triple:  lambda * exp(-x^1.5),  x = (d3-d1)/(w_pair+eps) >= 0
    const float xr = d3md1 * frcp(w_pair + EPS_F);
    const float triple = J_TRIPLE_LAMBDA * fexp2(-(xr * fsqrt(xr)) * L2E);
    const float w_eff = w_pair * (1.0f + bonus + triple);

    // softmin over b_j = 0.5*(d_j - d1) = 0.5*beta*(tk[0]-tk[j]), j=1..7.
    // List sorted, so the LSE max term is j=1:
    //   mm = -b1/tau = -(0.5*beta/tau)*(tk[0]-tk[1]) = -5*(tk[0]-tk[1])
    //   term_j - mm  = 5*(tk[j]-tk[1])
    const float KSM = 0.5f * BETA_F / SOFTMIN_TAU;  // 5
    const float mm  = -KSM * (tk[0] - tk[1]);
    float se = 0.0f;
#pragma unroll
    for (int j = 1; j < 8; ++j)
        se += fexp2(KSM * (tk[j] - tk[1]) * L2E);
    const float sdist = -SOFTMIN_TAU * (mm + flog2(se) * LN2_F);

    const float wall = sigm((0.5f * w_eff - sdist) / BETA_F);
    out[qblock + lane] = wall * (gs * rsum) * (hs * rsum);
}

// ---------------------------------------------------------------------------
extern "C" void kernel_launch(void* const* d_in, const int* in_sizes, int n_in,
                              void* d_out, int out_size, void* d_ws, size_t ws_size,
                              hipStream_t stream) {
    const float* uv         = (const float*)d_in[0];
    const float* seeds_raw  = (const float*)d_in[1];
    const float* w_raw      = (const float*)d_in[2];
    const float* h_raw      = (const float*)d_in[3];
    const float* theta      = (const float*)d_in[4];
    const float* a_raw      = (const float*)d_in[5];
    const float* seed_gates = (const float*)d_in[6];
    float* out = (float*)d_out;

    float* coeff = (float*)d_ws;             // 8*256 floats
    float* hg    = coeff + 8 * S_SEEDS;      // 2*256 floats (h/gate interleaved)

    voro_setup<<<1, S_SEEDS, 0, stream>>>(seeds_raw, h_raw, theta, a_raw,
                                          seed_gates, coeff, hg);

    const int q = in_sizes[0] / 2;           // number of query points
    voro_main<<<q / 32, 32, 0, stream>>>(uv, seeds_raw, w_raw, coeff, hg, out);
}